// PointMaskRCNN_21036749816356
// MI455X (gfx1250) — compile-verified
//
#include <hip/hip_runtime.h>
#include <stdint.h>

#define NB 2
#define NN 8192
#define NC 11
#define NW (NN / 32)          // 256 mask words per row
#define PI_F 3.14159265358979323846f
#define NMS_THR 0.3f
#define SCORE_THR 0.2f

typedef __attribute__((ext_vector_type(2))) float v2f;
typedef __attribute__((ext_vector_type(8))) float v8f;

// ---------------------------------------------------------------- kernel 1
// Per detection: argmax class (first max), max score, validity.
__global__ void k_classify(const float* __restrict__ cp,
                           int* __restrict__ cls, float* __restrict__ score,
                           uint8_t* __restrict__ valid) {
  int g = blockIdx.x * blockDim.x + threadIdx.x;
  if (g >= NB * NN) return;
  const float* p = cp + (size_t)g * NC;
  float best = p[0];
  int bi = 0;
#pragma unroll
  for (int c = 1; c < NC; ++c) {
    float v = p[c];
    if (v > best) { best = v; bi = c; }   // strict > keeps first max (jnp.argmax)
  }
  cls[g] = bi;
  score[g] = best;
  valid[g] = (bi > 0 && best >= SCORE_THR) ? 1 : 0;
}

// ---------------------------------------------------------------- kernel 2
// One block per batch: stable descending sort by score (bitonic in LDS on
// u64 keys = scorebits<<32 | (~idx), so ties break toward smaller idx),
// then gather sorted SoA.
__global__ void k_sort(const float* __restrict__ score,
                       const float* __restrict__ bsphere,
                       const int* __restrict__ cls,
                       const uint8_t* __restrict__ valid,
                       int* __restrict__ sIdx,
                       float* __restrict__ sX, float* __restrict__ sY,
                       float* __restrict__ sZ, float* __restrict__ sR,
                       float* __restrict__ sSQ, int* __restrict__ sCls,
                       uint8_t* __restrict__ sValid) {
  extern __shared__ unsigned long long key[];   // NN entries (64 KB dynamic LDS)
  const int b = blockIdx.x;
  const int tid = threadIdx.x;
  const int nt = blockDim.x;

  const float* sc = score + (size_t)b * NN;
  for (int i = tid; i < NN; i += nt) {
    unsigned int bits = __float_as_uint(sc[i]);   // scores are >= 0 -> order-preserving
    key[i] = ((unsigned long long)bits << 32) |
             (unsigned long long)(0xFFFFFFFFu - (unsigned int)i);
  }
  __syncthreads();

  for (int k = 2; k <= NN; k <<= 1) {
    for (int j = k >> 1; j > 0; j >>= 1) {
      for (int i = tid; i < NN; i += nt) {
        int ixj = i ^ j;
        if (ixj > i) {
          unsigned long long a = key[i], c = key[ixj];
          bool up = (i & k) == 0;                 // descending result overall
          bool sw = up ? (a < c) : (a > c);
          if (sw) { key[i] = c; key[ixj] = a; }
        }
      }
      __syncthreads();
    }
  }

  const float* bs = bsphere + (size_t)b * NN * 4;
  for (int s = tid; s < NN; s += nt) {
    unsigned long long kk = key[s];
    int idx = (int)(0xFFFFFFFFu - (unsigned int)(kk & 0xFFFFFFFFull));
    size_t o = (size_t)b * NN + s;
    sIdx[o] = idx;
    float x = bs[idx * 4 + 0], y = bs[idx * 4 + 1];
    float z = bs[idx * 4 + 2], r = bs[idx * 4 + 3];
    sX[o] = x; sY[o] = y; sZ[o] = z; sR[o] = r;
    sSQ[o] = x * x + y * y + z * z;
    sCls[o] = cls[(size_t)b * NN + idx];
    sValid[o] = valid[(size_t)b * NN + idx];
  }
}

// ---------------------------------------------------------------- kernel 3
// Sphere-IoU predicate from a WMMA-produced center dot product.
__device__ __forceinline__ bool pair_pred(float dot, float sqi, float ra, int ci,
                                          float sqj, float rb, int cj) {
  float d2 = sqi + sqj - 2.0f * dot;
  float dist = sqrtf(fmaxf(d2, 0.0f));
  float diff = fabsf(ra - rb);
  float sum = ra + rb;
  bool contained = dist <= diff;
  bool overlap = (dist > diff) && (dist < sum);
  float mn = fminf(ra, rb), mx = fmaxf(ra, rb);
  float q = mn / fmaxf(mx, 1e-12f);
  float iou_c = q * q * q;
  float sd = overlap ? dist : 1.0f;
  float t = sum - sd;
  float inter = t * t * (sd * sd + 2.0f * sd * sum - 3.0f * diff * diff) *
                (PI_F / (12.0f * sd));
  float uni = (4.0f / 3.0f) * PI_F * (ra * ra * ra + rb * rb * rb) - inter;
  float iou = contained ? iou_c : (overlap ? (inter / uni) : 0.0f);
  return (ci == cj) && (iou > NMS_THR);
}

// Each wave: 16 rows x 32 cols via two V_WMMA_F32_16X16X4_F32 Gram tiles,
// reduced to bit-packed suppression words with __ballot (wave32).
// Block = 8 waves -> 16 rows x 256 cols. Grid = (NN/256, NN/16, NB).
// Only the lower triangle (j < i) is ever consulted by the scan, so blocks /
// waves entirely above the diagonal (min j0 > i0+14) are skipped.
__global__ void k_mask(const float* __restrict__ sX, const float* __restrict__ sY,
                       const float* __restrict__ sZ, const float* __restrict__ sR,
                       const float* __restrict__ sSQ, const int* __restrict__ sCls,
                       unsigned int* __restrict__ mask) {
  const int batch = blockIdx.z;
  const int i0 = blockIdx.y * 16;
  const int j0blk = blockIdx.x * 256;
  if (j0blk > i0 + 14) return;               // whole block above diagonal: no barrier yet, safe

  const int lane = threadIdx.x & 31;
  const int wave = threadIdx.x >> 5;
  const int j0 = j0blk + wave * 32;
  const size_t base = (size_t)batch * NN;

  __shared__ float lsq[16], lr[16];
  __shared__ int lc[16];
  if (threadIdx.x < 16) {
    lsq[threadIdx.x] = sSQ[base + i0 + threadIdx.x];
    lr[threadIdx.x]  = sR [base + i0 + threadIdx.x];
    lc[threadIdx.x]  = sCls[base + i0 + threadIdx.x];
  }
  __syncthreads();                           // only barrier in this kernel
  if (j0 > i0 + 14) return;                  // wave above diagonal (post-barrier, safe)

  const int m = lane & 15;
  const bool hi = lane >= 16;

  // A (16x4 f32): lanes 0-15 hold K=0,1; lanes 16-31 hold K=2,3 (pad K=3 with 0)
  v2f a;
  a.x = hi ? sZ[base + i0 + m] : sX[base + i0 + m];
  a.y = hi ? 0.0f              : sY[base + i0 + m];

  const int c0 = j0 + m;
  const int c1 = j0 + 16 + m;
  v2f bb0, bb1;                     // B (4x16 f32), symmetric K split
  bb0.x = hi ? sZ[base + c0] : sX[base + c0];
  bb0.y = hi ? 0.0f          : sY[base + c0];
  bb1.x = hi ? sZ[base + c1] : sX[base + c1];
  bb1.y = hi ? 0.0f          : sY[base + c1];

  const float sq0 = sSQ[base + c0], r0 = sR[base + c0];
  const float sq1 = sSQ[base + c1], r1 = sR[base + c1];
  const int cc0 = sCls[base + c0], cc1 = sCls[base + c1];

  v8f acc = {};
  v8f d0 = __builtin_amdgcn_wmma_f32_16x16x4_f32(false, a, false, bb0,
                                                 (short)0, acc, false, false);
  v8f d1 = __builtin_amdgcn_wmma_f32_16x16x4_f32(false, a, false, bb1,
                                                 (short)0, acc, false, false);

  const int rowOff = hi ? 8 : 0;
  const int wIdx = j0 >> 5;
#pragma unroll
  for (int v = 0; v < 8; ++v) {
    const int row = rowOff + v;
    const float sqi = lsq[row], ri = lr[row];
    const int ci = lc[row];
    bool p0 = pair_pred(d0[v], sqi, ri, ci, sq0, r0, cc0);
    bool p1 = pair_pred(d1[v], sqi, ri, ci, sq1, r1, cc1);
    unsigned int u0 = (unsigned int)__ballot(p0);   // bits 0-15: M=v, 16-31: M=v+8
    unsigned int u1 = (unsigned int)__ballot(p1);
    unsigned int wlo = (u0 & 0xFFFFu) | (u1 << 16);          // row v, cols j0..j0+31
    unsigned int whi = (u0 >> 16) | (u1 & 0xFFFF0000u);      // row v+8
    if (lane == 0) {
      mask[((size_t)batch * NN + i0 + v)     * NW + wIdx] = wlo;
      mask[((size_t)batch * NN + i0 + 8 + v) * NW + wIdx] = whi;
    }
  }
}

// ---------------------------------------------------------------- kernel 4
// Sequential greedy NMS scan in sorted order. One 256-thread block per batch;
// thread t owns kept-word t in a register (8192 bits total). Row i+1 is
// prefetched before the barriers of iteration i to hide global latency.
__global__ void k_scan(const unsigned int* __restrict__ mask,
                       const uint8_t* __restrict__ sValid,
                       uint8_t* __restrict__ keepS) {
  const int b = blockIdx.x;
  const int t = threadIdx.x;       // 0..255
  const int lane = t & 31, wave = t >> 5;
  unsigned int kept = 0;
  __shared__ unsigned int wf[8];
  __shared__ int keepFlag;
  const unsigned int* mrow = mask + (size_t)b * NN * NW;
  const uint8_t* vs = sValid + (size_t)b * NN;
  uint8_t* ks = keepS + (size_t)b * NN;

  unsigned int mnext = mrow[t];    // row 0
  for (int i = 0; i < NN; ++i) {
    unsigned int mword = mnext;
    if (i + 1 < NN) mnext = mrow[(size_t)(i + 1) * NW + t];  // prefetch next row
    bool hit = (kept & mword) != 0;               // kept only has bits j < i
    unsigned long long bal = __ballot(hit);
    if (lane == 0) wf[wave] = (bal != 0) ? 1u : 0u;
    __syncthreads();
    if (t == 0) {
      unsigned int any = wf[0] | wf[1] | wf[2] | wf[3] |
                         wf[4] | wf[5] | wf[6] | wf[7];
      int k = (vs[i] != 0) && (any == 0);
      keepFlag = k;
      ks[i] = (uint8_t)k;
    }
    __syncthreads();
    if (keepFlag && t == (i >> 5)) kept |= (1u << (i & 31));
  }
}

// ---------------------------------------------------------------- kernel 5a
__global__ void k_scatter(const int* __restrict__ sIdx,
                          const uint8_t* __restrict__ keepS,
                          uint8_t* __restrict__ keepO) {
  int g = blockIdx.x * blockDim.x + threadIdx.x;
  if (g >= NB * NN) return;
  keepO[(size_t)(g / NN) * NN + sIdx[g]] = keepS[g];
}

// ---------------------------------------------------------------- kernel 5b
__global__ void k_final(const float* __restrict__ bs, const float* __restrict__ whl,
                        const float* __restrict__ yaw, const float* __restrict__ vel,
                        const int* __restrict__ cls, const float* __restrict__ score,
                        const uint8_t* __restrict__ keepO,
                        float* __restrict__ det, float* __restrict__ keepOut) {
  int g = blockIdx.x * blockDim.x + threadIdx.x;
  if (g >= NB * NN) return;
  float k = keepO[g] ? 1.0f : 0.0f;
  const float* b4 = bs + (size_t)g * 4;
  const float* w3 = whl + (size_t)g * 3;
  const float* v2 = vel + (size_t)g * 2;
  float* d = det + (size_t)g * 12;
  d[0] = b4[0] * k; d[1] = b4[1] * k; d[2] = b4[2] * k; d[3] = b4[3] * k;
  d[4] = (float)cls[g] * k; d[5] = score[g] * k;
  d[6] = w3[0] * k; d[7] = w3[1] * k; d[8] = w3[2] * k;
  d[9] = yaw[g] * k;
  d[10] = v2[0] * k; d[11] = v2[1] * k;
  keepOut[g] = k;
}

// ---------------------------------------------------------------- launch
extern "C" void kernel_launch(void* const* d_in, const int* in_sizes, int n_in,
                              void* d_out, int out_size, void* d_ws, size_t ws_size,
                              hipStream_t stream) {
  (void)in_sizes; (void)n_in; (void)out_size; (void)ws_size;
  const float* cp  = (const float*)d_in[0];   // [2,8192,11]
  const float* bs  = (const float*)d_in[1];   // [2,8192,4]
  const float* whl = (const float*)d_in[2];   // [2,8192,3]
  const float* yaw = (const float*)d_in[3];   // [2,8192,1]
  const float* vel = (const float*)d_in[4];   // [2,8192,2]

  float* out = (float*)d_out;
  float* det = out;                            // 2*8192*12
  float* keepOut = out + (size_t)NB * NN * 12; // 2*8192

  // workspace bump allocator (total ~16.6 MB)
  char* w = (char*)d_ws;
  auto take = [&](size_t bytes) -> char* {
    char* p = w;
    w += (bytes + 255) & ~(size_t)255;
    return p;
  };
  const size_t nBN = (size_t)NB * NN;
  int*      cls    = (int*)take(nBN * sizeof(int));
  float*    score  = (float*)take(nBN * sizeof(float));
  uint8_t*  valid  = (uint8_t*)take(nBN);
  int*      sIdx   = (int*)take(nBN * sizeof(int));
  float*    sX     = (float*)take(nBN * sizeof(float));
  float*    sY     = (float*)take(nBN * sizeof(float));
  float*    sZ     = (float*)take(nBN * sizeof(float));
  float*    sR     = (float*)take(nBN * sizeof(float));
  float*    sSQ    = (float*)take(nBN * sizeof(float));
  int*      sCls   = (int*)take(nBN * sizeof(int));
  uint8_t*  sValid = (uint8_t*)take(nBN);
  uint8_t*  keepS  = (uint8_t*)take(nBN);
  uint8_t*  keepO  = (uint8_t*)take(nBN);
  unsigned int* mask = (unsigned int*)take(nBN * NW * sizeof(unsigned int)); // 16 MB

  k_classify<<<(NB * NN + 255) / 256, 256, 0, stream>>>(cp, cls, score, valid);
  k_sort<<<NB, 1024, NN * sizeof(unsigned long long), stream>>>(
      score, bs, cls, valid, sIdx, sX, sY, sZ, sR, sSQ, sCls, sValid);
  dim3 g3(NN / 256, NN / 16, NB);
  k_mask<<<g3, 256, 0, stream>>>(sX, sY, sZ, sR, sSQ, sCls, mask);
  k_scan<<<NB, 256, 0, stream>>>(mask, sValid, keepS);
  k_scatter<<<(NB * NN + 255) / 256, 256, 0, stream>>>(sIdx, keepS, keepO);
  k_final<<<(NB * NN + 255) / 256, 256, 0, stream>>>(bs, whl, yaw, vel, cls, score,
                                                     keepO, det, keepOut);
}